// PointGroup_3616362463480
// MI455X (gfx1250) — compile-verified
//
#include <hip/hip_runtime.h>
#include <stdint.h>

// Problem constants (match reference harness)
#define NPTS   1000000
#define CFEAT  32
#define NCLUST 512
#define PPC    4096
#define STOT   (NCLUST*PPC)        // 2,097,152
#define FS     14
#define FS3    (FS*FS*FS)          // 2744
#define NVOX   (NCLUST*FS3)        // 1,404,928

// workspace layout (32-bit word offsets)
#define WS_VCOUNT 0
#define WS_VID    (NVOX)
#define WS_CSUM   (WS_VID + STOT)
#define WS_CCNT   (WS_CSUM + NCLUST*3)
#define WS_CMIN   (WS_CCNT + NCLUST)
#define WS_CMAX   (WS_CMIN + NCLUST*3)
#define WS_CPAR   (WS_CMAX + NCLUST*3)

// ---- order-preserving float<->uint mapping for atomic min/max ----
__device__ __forceinline__ unsigned fmap(float f) {
  unsigned b = __float_as_uint(f);
  return (b & 0x80000000u) ? ~b : (b | 0x80000000u);
}
__device__ __forceinline__ float funmap(unsigned u) {
  unsigned b = (u & 0x80000000u) ? (u & 0x7FFFFFFFu) : ~u;
  return __uint_as_float(b);
}

// ---- CDNA5 async global->LDS DMA (ASYNCcnt pipeline) ----
__device__ __forceinline__ void async_ld_b32(unsigned lds_off, unsigned voff,
                                             const int* base) {
  asm volatile("global_load_async_to_lds_b32 %0, %1, %2"
               :: "v"(lds_off), "v"(voff), "s"(base) : "memory");
}
__device__ __forceinline__ void wait_async_le2() {
  asm volatile("s_wait_asynccnt 2" ::: "memory");
}
__device__ __forceinline__ void wait_async_le0() {
  asm volatile("s_wait_asynccnt 0" ::: "memory");
}

// ------------------------------------------------------------------
// vectorized zero: n4 = number of float4 elements
__global__ void k_zero4(float4* __restrict__ p, int n4) {
  int g = blockIdx.x * blockDim.x + threadIdx.x;
  if (g < n4) p[g] = make_float4(0.f, 0.f, 0.f, 0.f);
}

__global__ void k_init_stats(float* __restrict__ csum, float* __restrict__ ccnt,
                             unsigned* __restrict__ cminb, unsigned* __restrict__ cmaxb) {
  int i = blockIdx.x * blockDim.x + threadIdx.x;
  if (i < NCLUST*3) { csum[i] = 0.0f; cminb[i] = 0xFFFFFFFFu; cmaxb[i] = 0u; }
  if (i < NCLUST)   ccnt[i] = 0.0f;
}

// ---- pass 1: per-cluster sum/count/min/max of raw gathered coords ----
__global__ void k_cluster_stats(const float* __restrict__ coords,
                                const int* __restrict__ idx,
                                const int* __restrict__ seg,
                                float* __restrict__ csum, float* __restrict__ ccnt,
                                unsigned* __restrict__ cminb, unsigned* __restrict__ cmaxb) {
  __shared__ float    s_sum[NCLUST*3];
  __shared__ float    s_cnt[NCLUST];
  __shared__ unsigned s_min[NCLUST*3];
  __shared__ unsigned s_max[NCLUST*3];
  for (int i = threadIdx.x; i < NCLUST*3; i += blockDim.x) {
    s_sum[i] = 0.0f; s_min[i] = 0xFFFFFFFFu; s_max[i] = 0u;
  }
  for (int i = threadIdx.x; i < NCLUST; i += blockDim.x) s_cnt[i] = 0.0f;
  __syncthreads();

  const int chunk = STOT / gridDim.x;          // 4096 with 512 blocks
  const int base  = blockIdx.x * chunk;
  const int lane  = threadIdx.x & 31;

  for (int j = threadIdx.x; j < chunk; j += blockDim.x) {
    int s = base + j;
    __builtin_prefetch(&idx[s + 1024], 0, 0);   // global_prefetch_b8
    int p = idx[s];
    int c = seg[s];
    float x = coords[3*p+0], y = coords[3*p+1], z = coords[3*p+2];

    int c0 = __shfl(c, 0);
    if (__all(c == c0)) {
      // wave-uniform cluster: wave32 shuffle reduction, one LDS atomic set per wave
      float sx=x, sy=y, sz=z, nx=x, ny=y, nz=z, mx=x, my=y, mz=z;
      for (int off = 16; off >= 1; off >>= 1) {
        sx += __shfl_down(sx, off); sy += __shfl_down(sy, off); sz += __shfl_down(sz, off);
        nx = fminf(nx, __shfl_down(nx, off));
        ny = fminf(ny, __shfl_down(ny, off));
        nz = fminf(nz, __shfl_down(nz, off));
        mx = fmaxf(mx, __shfl_down(mx, off));
        my = fmaxf(my, __shfl_down(my, off));
        mz = fmaxf(mz, __shfl_down(mz, off));
      }
      if (lane == 0) {
        atomicAdd(&s_sum[3*c0+0], sx);
        atomicAdd(&s_sum[3*c0+1], sy);
        atomicAdd(&s_sum[3*c0+2], sz);
        atomicAdd(&s_cnt[c0], 32.0f);
        atomicMin(&s_min[3*c0+0], fmap(nx));
        atomicMin(&s_min[3*c0+1], fmap(ny));
        atomicMin(&s_min[3*c0+2], fmap(nz));
        atomicMax(&s_max[3*c0+0], fmap(mx));
        atomicMax(&s_max[3*c0+1], fmap(my));
        atomicMax(&s_max[3*c0+2], fmap(mz));
      }
    } else {
      atomicAdd(&s_sum[3*c+0], x);
      atomicAdd(&s_sum[3*c+1], y);
      atomicAdd(&s_sum[3*c+2], z);
      atomicAdd(&s_cnt[c], 1.0f);
      atomicMin(&s_min[3*c+0], fmap(x));
      atomicMin(&s_min[3*c+1], fmap(y));
      atomicMin(&s_min[3*c+2], fmap(z));
      atomicMax(&s_max[3*c+0], fmap(x));
      atomicMax(&s_max[3*c+1], fmap(y));
      atomicMax(&s_max[3*c+2], fmap(z));
    }
  }
  __syncthreads();
  for (int c = threadIdx.x; c < NCLUST; c += blockDim.x) {
    float cnt = s_cnt[c];
    if (cnt > 0.0f) {
      atomicAdd(&ccnt[c], cnt);
      for (int d = 0; d < 3; ++d) {
        atomicAdd(&csum[3*c+d], s_sum[3*c+d]);
        atomicMin(&cminb[3*c+d], s_min[3*c+d]);
        atomicMax(&cmaxb[3*c+d], s_max[3*c+d]);
      }
    }
  }
}

// ---- pass 2: per-cluster transform parameters + center/size outputs ----
__global__ void k_cluster_params(const float* __restrict__ csum, const float* __restrict__ ccnt,
                                 const unsigned* __restrict__ cminb, const unsigned* __restrict__ cmaxb,
                                 const float* __restrict__ rand3a, const float* __restrict__ rand3b,
                                 const int* __restrict__ fsp, const int* __restrict__ sclp,
                                 float* __restrict__ cpar,
                                 float* __restrict__ out_center, float* __restrict__ out_size) {
  int c = blockIdx.x * blockDim.x + threadIdx.x;
  if (c >= NCLUST) return;
  float fsf = (float)fsp[0];
  float scl = (float)sclp[0];
  float cnt = ccnt[c];
  float mean[3], cmn[3], cmx[3];
  float mrel = -1e30f;
  for (int d = 0; d < 3; ++d) {
    mean[d] = csum[3*c+d] / cnt;
    float rmin = funmap(cminb[3*c+d]);
    float rmax = funmap(cmaxb[3*c+d]);
    cmn[d] = rmin - mean[d];                  // min of centered == rawmin - mean
    cmx[d] = rmax - mean[d];
    out_size[3*c+d]   = cmx[d] - cmn[d];
    out_center[3*c+d] = (cmx[d] + cmn[d]) * 0.5f + mean[d];
    mrel = fmaxf(mrel, (cmx[d] - cmn[d]) / fsf);
  }
  float cscale = 1.0f / mrel - 0.01f;
  cscale = fminf(cscale, scl);
  for (int d = 0; d < 3; ++d) {
    float mn  = cmn[d] * cscale;
    float mx  = cmx[d] * cscale;
    float rng = mx - mn;
    float off = -mn + fmaxf(fsf - rng - 0.001f, 0.0f) * rand3a[d]
                    + fminf(fsf - rng + 0.001f, 0.0f) * rand3b[d];
    cpar[8*c+d]   = mean[d];
    cpar[8*c+4+d] = off;
  }
  cpar[8*c+3] = cscale;
  cpar[8*c+7] = 0.0f;
}

// ---- pass 3: voxel id per point + voxel counts (async LDS-staged indices) ----
__global__ void k_voxelize(const float* __restrict__ coords,
                           const int* __restrict__ idx,
                           const int* __restrict__ seg,
                           const float* __restrict__ cpar,
                           int* __restrict__ vid,
                           float* __restrict__ vcount) {
  __shared__ int sh_idx[2][256];
  __shared__ int sh_seg[2][256];
  const int TILE = 256, TILES = 8;
  const int base = blockIdx.x * (TILE * TILES);
  const int tid  = threadIdx.x;

  unsigned lds_i[2], lds_s[2];
  lds_i[0] = (unsigned)(uintptr_t)&sh_idx[0][tid];
  lds_i[1] = (unsigned)(uintptr_t)&sh_idx[1][tid];
  lds_s[0] = (unsigned)(uintptr_t)&sh_seg[0][tid];
  lds_s[1] = (unsigned)(uintptr_t)&sh_seg[1][tid];

  // prologue: async-DMA tile 0 indices into LDS
  {
    unsigned off0 = (unsigned)((base + tid) * 4);
    async_ld_b32(lds_i[0], off0, idx);
    async_ld_b32(lds_s[0], off0, seg);
  }

  for (int t = 0; t < TILES; ++t) {
    int buf = t & 1;
    if (t + 1 < TILES) {
      unsigned off = (unsigned)((base + (t+1)*TILE + tid) * 4);
      async_ld_b32(lds_i[buf^1], off, idx);
      async_ld_b32(lds_s[buf^1], off, seg);
      wait_async_le2();                    // tile t's 2 loads complete (in-order)
    } else {
      wait_async_le0();
    }
    int p = sh_idx[buf][tid];
    int c = sh_seg[buf][tid];
    const float* cp = &cpar[c*8];
    float m0 = cp[0], m1 = cp[1], m2 = cp[2], cs = cp[3];
    float o0 = cp[4], o1 = cp[5], o2 = cp[6];
    // exact reference op order: ((raw - mean) * cscale) + off
    float x = (coords[3*p+0] - m0) * cs + o0;
    float y = (coords[3*p+1] - m1) * cs + o1;
    float z = (coords[3*p+2] - m2) * cs + o2;
    int vx = (int)floorf(x); vx = vx < 0 ? 0 : (vx > FS-1 ? FS-1 : vx);
    int vy = (int)floorf(y); vy = vy < 0 ? 0 : (vy > FS-1 ? FS-1 : vy);
    int vz = (int)floorf(z); vz = vz < 0 ? 0 : (vz > FS-1 ? FS-1 : vz);
    int v = ((c*FS + vx)*FS + vy)*FS + vz;
    vid[base + t*TILE + tid] = v;
    atomicAdd(&vcount[v], 1.0f);
  }
}

// ---- pass 4: scatter-add features, one thread per (point, 4 channels) ----
__global__ void k_scatter_feats(const float4* __restrict__ feats4,
                                const int* __restrict__ idx,
                                const int* __restrict__ vid,
                                float* __restrict__ vfeats) {
  int g  = blockIdx.x * blockDim.x + threadIdx.x;  // < STOT*CFEAT/4 = 2^24
  int s  = g >> 3;                                 // point (8 lanes share it)
  int q  = g & 7;                                  // float4 chunk within 32 channels
  int p  = idx[s];
  int v  = vid[s];
  float4 f = feats4[p * (CFEAT/4) + q];            // coalesced b128 gather (L2-resident)
  float* dst = &vfeats[v * CFEAT + q * 4];
  atomicAdd(dst + 0, f.x);
  atomicAdd(dst + 1, f.y);
  atomicAdd(dst + 2, f.z);
  atomicAdd(dst + 3, f.w);
}

// ---- pass 5: mean-mode divide (float4 per thread) ----
__global__ void k_finalize(float4* __restrict__ vfeats4, const float* __restrict__ vcount) {
  int g = blockIdx.x * blockDim.x + threadIdx.x;   // < NVOX*CFEAT/4
  int v = g >> 3;                                  // voxel (8 lanes share it)
  float d = fmaxf(vcount[v], 1.0f);
  float4 f = vfeats4[g];
  f.x = f.x / d; f.y = f.y / d; f.z = f.z / d; f.w = f.w / d;
  vfeats4[g] = f;
}

// ------------------------------------------------------------------
extern "C" void kernel_launch(void* const* d_in, const int* in_sizes, int n_in,
                              void* d_out, int out_size, void* d_ws, size_t ws_size,
                              hipStream_t stream) {
  (void)in_sizes; (void)n_in; (void)ws_size;
  const float*  feats  = (const float*)d_in[0];
  const float*  coords = (const float*)d_in[1];
  const int*    idx    = (const int*)d_in[2];
  const int*    seg    = (const int*)d_in[3];
  const float*  r3a    = (const float*)d_in[4];
  const float*  r3b    = (const float*)d_in[5];
  const int*    fsp    = (const int*)d_in[6];
  const int*    sclp   = (const int*)d_in[7];

  float* out        = (float*)d_out;
  float* vfeats     = out;
  float* out_center = out + (size_t)NVOX * CFEAT;
  float* out_sizep  = out_center + NCLUST * 3;

  float*    ws     = (float*)d_ws;
  float*    vcount = ws + WS_VCOUNT;
  int*      vid    = (int*)(ws + WS_VID);
  float*    csum   = ws + WS_CSUM;
  float*    ccnt   = ws + WS_CCNT;
  unsigned* cminb  = (unsigned*)(ws + WS_CMIN);
  unsigned* cmaxb  = (unsigned*)(ws + WS_CMAX);
  float*    cpar   = ws + WS_CPAR;

  // out_size (44,960,768) and NVOX are divisible by 4
  int out4 = out_size / 4;
  k_zero4<<<(out4 + 255) / 256, 256, 0, stream>>>((float4*)out, out4);
  k_zero4<<<(NVOX/4 + 255) / 256, 256, 0, stream>>>((float4*)vcount, NVOX/4);
  k_init_stats<<<(NCLUST*3 + 255) / 256, 256, 0, stream>>>(csum, ccnt, cminb, cmaxb);

  k_cluster_stats<<<512, 256, 0, stream>>>(coords, idx, seg, csum, ccnt, cminb, cmaxb);
  k_cluster_params<<<2, 256, 0, stream>>>(csum, ccnt, cminb, cmaxb, r3a, r3b,
                                          fsp, sclp, cpar, out_center, out_sizep);
  k_voxelize<<<STOT / (256*8), 256, 0, stream>>>(coords, idx, seg, cpar, vid, vcount);
  k_scatter_feats<<<(STOT * (CFEAT/4)) / 256, 256, 0, stream>>>((const float4*)feats, idx, vid, vfeats);
  k_finalize<<<(NVOX * (CFEAT/4)) / 256, 256, 0, stream>>>((float4*)vfeats, vcount);
}